// PhysicalTensorICTDRecovery_26628797235368
// MI455X (gfx1250) — compile-verified
//
#include <hip/hip_runtime.h>

// ---------------------------------------------------------------------------
// PhysicalTensorICTDRecovery for MI455X (gfx1250, wave32).
//
// Per row r = b*8+c (2.1M rows):
//   t[d]   = sum_m Wc[d,m] * xin[r,m]       (15x15 contraction, padded 16x16)
//   out[j] = t[dmap[j]] * (1/coef[j])       (j in [0,81))
//
// Store-bandwidth bound: ~680 MB out + ~126 MB in -> ~35 us floor @ 23.3 TB/s.
// Contraction on V_WMMA_F32_16X16X4_F32 (4 chained K=4 steps per wave).
// dmap/1/coef come from a compile-time __constant__ table (one b64 load per
// lane per chunk), so the expansion hot loop is ds_load + v_mul + NT store.
// ---------------------------------------------------------------------------

typedef __attribute__((ext_vector_type(2))) float v2f;
typedef __attribute__((ext_vector_type(8))) float v8f;

#define CCH        8      // channels
#define DDIM       15     // monomial dim
#define NOUT       81     // 3^4 cartesian entries
#define XCOLS      200    // x row width
#define ROWS_WG    128    // rows per workgroup (16 batch elems * 8 channels)
#define THREADS    256    // 8 waves
#define XSTRIDE    17     // padded LDS row stride (bank-conflict avoidance)

struct DmIc { int dm; float ic; };

// compile-time builder for the (dmap, 1/coef) table entry at cartesian index j
constexpr DmIc mk_dmic(int j) {
    const int g0 = j % 3, g1 = (j / 3) % 3, g2 = (j / 9) % 3, g3 = (j / 27) % 3;
    const int a  = (g0 == 0) + (g1 == 0) + (g2 == 0) + (g3 == 0);
    const int b  = (g0 == 1) + (g1 == 1) + (g2 == 1) + (g3 == 1);
    const int c  = 4 - a - b;
    const int   offs[5] = {10, 6, 3, 1, 0};           // monomial block offset by a
    const float fact[5] = {1.f, 1.f, 2.f, 6.f, 24.f};
    return DmIc{offs[a] + (4 - a - b), fact[a] * fact[b] * fact[c] / 24.0f};
}

#define E1(j)  mk_dmic(j)
#define E4(j)  E1(j), E1((j) + 1), E1((j) + 2), E1((j) + 3)
#define E16(j) E4(j), E4((j) + 4), E4((j) + 8), E4((j) + 12)
__constant__ DmIc DMIC[96] = { E16(0), E16(16), E16(32), E16(48), E16(64), E16(80) };
// entries j >= 81 are padding (valid data, never used)

__global__ __launch_bounds__(THREADS)
void ictd_wmma_kernel(const float* __restrict__ x,
                      const float* __restrict__ V4,
                      const float* __restrict__ V2,
                      const float* __restrict__ V0,
                      float* __restrict__ out)
{
    __shared__ float lds_x[ROWS_WG * XSTRIDE];  // input tile, 16 feats/row (col 15 = 0)
    __shared__ float lds_t[ROWS_WG * XSTRIDE];  // t results, 16 d/row
    __shared__ float lds_w[16 * 16];            // Wc[d][m], zero padded

    const int tid  = threadIdx.x;
    const int wg   = blockIdx.x;
    const int b0   = wg * (ROWS_WG / CCH);      // first batch index of this WG
    const int wave = tid >> 5;
    const int lane = tid & 31;
    const int half = lane >> 4;                 // 0: lanes 0-15, 1: lanes 16-31
    const int l16  = lane & 15;
    const int rbase = wave * 16;                // first local row of this wave

    // ---- per-lane (dmap, 1/coef) for j = chunk*32 + lane: 3x b64 const load --
    DmIc dmic[3];
#pragma unroll
    for (int cnk = 0; cnk < 3; ++cnk)
        dmic[cnk] = DMIC[cnk * 32 + lane];

    // ---- fused weight matrix Wc[d][m] (one element per thread) --------------
    {
        int d = tid >> 4, m = tid & 15;
        float w = 0.0f;
        if (d < DDIM) {
            if (m < 9)        w = V4[d * 9 + m];
            else if (m < 14)  w = V2[d * 5 + (m - 9)];
            else if (m == 14) w = V0[d];
        }
        lds_w[tid] = w;
    }

    // ---- stage used x columns -> 128x16 LDS tile (coalesced per segment) ----
    // l=4: cols [128,200) -> m = 0..8 ;  l=2: cols [32,72) -> m = 9..13 ;
    // l=0: cols [0,8)     -> m = 14   ;  m = 15 zero-padded.
    for (int e = tid; e < 16 * 72; e += THREADS) {
        int bl = e / 72, off = e % 72;
        float v = __builtin_nontemporal_load(&x[(size_t)(b0 + bl) * XCOLS + 128 + off]);
        int c = off / 9, m = off % 9;
        lds_x[(bl * CCH + c) * XSTRIDE + m] = v;
    }
    for (int e = tid; e < 16 * 40; e += THREADS) {
        int bl = e / 40, off = e % 40;
        float v = __builtin_nontemporal_load(&x[(size_t)(b0 + bl) * XCOLS + 32 + off]);
        int c = off / 5, m = 9 + off % 5;
        lds_x[(bl * CCH + c) * XSTRIDE + m] = v;
    }
    for (int e = tid; e < 16 * 8; e += THREADS) {
        int bl = e / 8, c = e % 8;
        float v = __builtin_nontemporal_load(&x[(size_t)(b0 + bl) * XCOLS + c]);
        lds_x[(bl * CCH + c) * XSTRIDE + 14] = v;
        lds_x[(bl * CCH + c) * XSTRIDE + 15] = 0.0f;    // K pad
    }
    __syncthreads();

    // ---- per-wave 16x16 tile: t = A(16x16) x Wc^T(16x16) via 4x WMMA K=4 ----
#if __has_builtin(__builtin_amdgcn_wmma_f32_16x16x4_f32)
    v8f acc = {};
#pragma unroll
    for (int k0 = 0; k0 < 16; k0 += 4) {
        // A 16x4 f32 layout: lanes 0-15 M=0..15 {K=k0,k0+1}; lanes 16-31 {K=k0+2,k0+3}
        const int kk = k0 + 2 * half;
        v2f a, bm;
        a.x  = lds_x[(rbase + l16) * XSTRIDE + kk];
        a.y  = lds_x[(rbase + l16) * XSTRIDE + kk + 1];
        // B 4x16 f32: B[K,N] = Wc[N,K]; N = lane%16, same K split as A
        bm.x = lds_w[l16 * 16 + kk];
        bm.y = lds_w[l16 * 16 + kk + 1];
        acc  = __builtin_amdgcn_wmma_f32_16x16x4_f32(false, a, false, bm,
                                                     (short)0, acc, false, false);
    }
    // D 16x16 f32 layout: VGPR g -> M = g + 8*half, N = l16
#pragma unroll
    for (int g = 0; g < 8; ++g) {
        lds_t[(rbase + g + 8 * half) * XSTRIDE + l16] = acc[g];
    }
#else
    // Scalar fallback (keeps the kernel compiling if the f32 WMMA builtin
    // is unavailable; histogram will show wmma==0 in that case).
    for (int rl = tid; rl < ROWS_WG; rl += THREADS) {
        for (int d = 0; d < 16; ++d) {
            float s = 0.0f;
            for (int m = 0; m < 16; ++m)
                s += lds_x[rl * XSTRIDE + m] * lds_w[d * 16 + m];
            lds_t[rl * XSTRIDE + d] = s;
        }
    }
#endif
    __syncthreads();

    // ---- expand 15 -> 81 and stream out (wave-owned rows, register tables) --
    // Wave w writes rows [w*16, w*16+16); lane covers j = chunk*32 + lane.
    // Hot loop per element: 1 ds_load (imm offset) + 1 v_mul + 1 NT store.
    const size_t out_base = (size_t)wg * (ROWS_WG * NOUT)
                          + (size_t)rbase * NOUT;
#pragma unroll
    for (int cnk = 0; cnk < 3; ++cnk) {
        const int j = cnk * 32 + lane;
        if (j < NOUT) {
            const float* tptr = &lds_t[rbase * XSTRIDE + dmic[cnk].dm];
            float* outp = out + out_base + j;
            const float s = dmic[cnk].ic;
#pragma unroll
            for (int r = 0; r < 16; ++r) {
                __builtin_nontemporal_store(tptr[r * XSTRIDE] * s, outp + r * NOUT);
            }
        }
    }
}

extern "C" void kernel_launch(void* const* d_in, const int* in_sizes, int n_in,
                              void* d_out, int out_size, void* d_ws, size_t ws_size,
                              hipStream_t stream)
{
    const float* x  = (const float*)d_in[0];   // (B, 200)
    const float* V4 = (const float*)d_in[1];   // (15, 9)
    const float* V2 = (const float*)d_in[2];   // (15, 5)
    const float* V0 = (const float*)d_in[3];   // (15, 1)
    float* out = (float*)d_out;                // (B, 8, 81)

    const int B    = in_sizes[0] / XCOLS;      // 262144
    const int rows = B * CCH;                  // 2,097,152 (divisible by 128)
    const int grid = rows / ROWS_WG;           // 16384 workgroups

    ictd_wmma_kernel<<<dim3(grid), dim3(THREADS), 0, stream>>>(x, V4, V2, V0, out);
}